// PromptPool_19164144075399
// MI455X (gfx1250) — compile-verified
//
#include <hip/hip_runtime.h>
#include <math.h>

#define HIDDEN   1024
#define POOL     512
#define BATCH    1024
#define TOPK     8
#define PRE_SEQ  32
#define EPS      1e-8f

typedef float v2f __attribute__((ext_vector_type(2)));
typedef float v4f __attribute__((ext_vector_type(4)));
typedef float v8f __attribute__((ext_vector_type(8)));

// ---------------------------------------------------------------------------
// Kernel 1: reciprocal L2 norms for Q rows (1024) and K rows (512).
// One wave32 per row; coalesced strided loads; butterfly reduction.
// ---------------------------------------------------------------------------
__global__ void __launch_bounds__(256)
norms_kernel(const float* __restrict__ q, const float* __restrict__ pk,
             float* __restrict__ rq, float* __restrict__ rk) {
    int wave = (blockIdx.x * blockDim.x + threadIdx.x) >> 5;
    int lane = threadIdx.x & 31;
    const float* src;
    float* dst;
    if (wave < BATCH) {
        src = q + (size_t)wave * HIDDEN;
        dst = rq + wave;
    } else {
        int r = wave - BATCH;
        if (r >= POOL) return;
        src = pk + (size_t)r * HIDDEN;
        dst = rk + r;
    }
    float s = 0.0f;
    #pragma unroll 8
    for (int i = lane; i < HIDDEN; i += 32) {
        float v = src[i];
        s = fmaf(v, v, s);
    }
    #pragma unroll
    for (int off = 16; off; off >>= 1) s += __shfl_xor(s, off, 32);
    if (lane == 0) *dst = 1.0f / fmaxf(sqrtf(s), EPS);
}

// ---------------------------------------------------------------------------
// Kernel 2: S[b][p] = (q_b . k_p) * rq[b] * rk[p] via V_WMMA_F32_16X16X4_F32.
// One wave per 16x16 output tile. A-fragment layout (ISA 7.12.2, 32-bit A
// 16x4): lanes 0-15 hold K=0,1 in v[0],v[1]; lanes 16-31 hold K=2,3.
// B = K^T, so the B-fragment load is the mirrored contiguous float2 from the
// prompts_key row. C layout: VGPR r -> M = r + 8*(lane>=16), N = lane&15.
// ---------------------------------------------------------------------------
__global__ void __launch_bounds__(256)
sim_wmma_kernel(const float* __restrict__ Q, const float* __restrict__ Km,
                const float* __restrict__ rq, const float* __restrict__ rk,
                float* __restrict__ S) {
    int wave = (blockIdx.x * blockDim.x + threadIdx.x) >> 5;
    int lane = threadIdx.x & 31;
    int tileM = wave >> 5;   // 64 tiles of 16 rows  (BATCH/16)
    int tileN = wave & 31;   // 32 tiles of 16 cols  (POOL/16)

    int mrow = tileM * 16 + (lane & 15);
    int nrow = tileN * 16 + (lane & 15);
    int koff = (lane >> 4) * 2;            // upper half-wave holds K=2,3

    const float* qrow = Q  + (size_t)mrow * HIDDEN + koff;
    const float* krow = Km + (size_t)nrow * HIDDEN + koff;

    v8f c = {};
    #pragma unroll 4
    for (int k = 0; k < HIDDEN; k += 4) {
        v2f a = *(const v2f*)(qrow + k);
        v2f b = *(const v2f*)(krow + k);
        c = __builtin_amdgcn_wmma_f32_16x16x4_f32(
                /*neg_a=*/false, a, /*neg_b=*/false, b,
                /*c_mod=*/(short)0, c, /*reuse_a=*/false, /*reuse_b=*/false);
    }

    int col  = tileN * 16 + (lane & 15);
    int half = (lane >> 4) * 8;
    float sk = rk[col];
    #pragma unroll
    for (int r = 0; r < 8; ++r) {
        int row = tileM * 16 + half + r;
        S[(size_t)row * POOL + col] = c[r] * rq[row] * sk;
    }
}

// ---------------------------------------------------------------------------
// Kernel 3: per-row top-8 (descending, lowest-index tie-break like torch).
// One wave per row: 16 register values per lane, 8 rounds of wave max-reduce.
// ---------------------------------------------------------------------------
__global__ void __launch_bounds__(256)
topk_kernel(const float* __restrict__ S, float* __restrict__ top_out,
            int* __restrict__ idx_out) {
    int wave = (blockIdx.x * blockDim.x + threadIdx.x) >> 5;
    int lane = threadIdx.x & 31;
    if (wave >= BATCH) return;

    const float* row = S + (size_t)wave * POOL;
    float vals[16];
    #pragma unroll
    for (int i = 0; i < 16; ++i) vals[i] = row[lane + 32 * i];

    for (int t = 0; t < TOPK; ++t) {
        // local max over 16 slots (ascending slot => earliest/lowest col kept)
        float bv = vals[0];
        int bslot = 0;
        #pragma unroll
        for (int i = 1; i < 16; ++i)
            if (vals[i] > bv) { bv = vals[i]; bslot = i; }
        int bi = lane + 32 * bslot;

        // wave-wide argmax, tie -> lower index
        #pragma unroll
        for (int off = 16; off; off >>= 1) {
            float ov = __shfl_xor(bv, off, 32);
            int   oi = __shfl_xor(bi, off, 32);
            if (ov > bv || (ov == bv && oi < bi)) { bv = ov; bi = oi; }
        }

        if (lane == 0) {
            top_out[wave * TOPK + t] = bv;
            idx_out[wave * TOPK + t] = bi;
        }
        if (lane == (bi & 31)) vals[bi >> 5] = -INFINITY;  // knock out winner
    }
}

// ---------------------------------------------------------------------------
// Kernel 4: gather prompts[idx] -> out. One block per (b,k): 32*1024 floats =
// 8192 x 128-bit. Pool (64 MB) stays L2-resident via regular-temporal loads;
// the 1 GiB output uses non-temporal stores so it doesn't thrash L2.
// ---------------------------------------------------------------------------
__global__ void __launch_bounds__(256)
gather_kernel(const float* __restrict__ prompts, const int* __restrict__ idx,
              float* __restrict__ out) {
    int bk = blockIdx.x;                       // 0 .. BATCH*TOPK-1
    int p  = idx[bk];
    const v4f* __restrict__ src =
        (const v4f*)(prompts + (size_t)p * (PRE_SEQ * HIDDEN));
    v4f* __restrict__ dst = (v4f*)(out + (size_t)bk * (PRE_SEQ * HIDDEN));
    int tid = threadIdx.x;
    #pragma unroll
    for (int i = 0; i < 32; ++i) {
        v4f v = src[tid + 256 * i];
        __builtin_nontemporal_store(v, &dst[tid + 256 * i]);
    }
}

// ---------------------------------------------------------------------------
extern "C" void kernel_launch(void* const* d_in, const int* in_sizes, int n_in,
                              void* d_out, int out_size, void* d_ws, size_t ws_size,
                              hipStream_t stream) {
    const float* querys  = (const float*)d_in[0];   // [1024, 1024]
    const float* pkeys   = (const float*)d_in[1];   // [512, 1024]
    const float* prompts = (const float*)d_in[2];   // [512, 32, 1024]
    float* out = (float*)d_out;

    // workspace layout (floats): rq[1024] | rk[512] | S[1024*512] | idx[8192]
    float* ws  = (float*)d_ws;
    float* rq  = ws;
    float* rk  = ws + BATCH;
    float* S   = ws + BATCH + POOL;
    int*   idx = (int*)(ws + BATCH + POOL + (size_t)BATCH * POOL);

    // output: gathered_prompts [1024,8,32,1024] then top_sim [1024,8]
    float* top_sim = out + (size_t)BATCH * TOPK * PRE_SEQ * HIDDEN;

    // 1536 rows, 1 wave each, 8 waves/block -> 192 blocks
    norms_kernel<<<192, 256, 0, stream>>>(querys, pkeys, rq, rk);

    // 64*32 = 2048 tiles, 8 waves/block -> 256 blocks
    sim_wmma_kernel<<<256, 256, 0, stream>>>(querys, pkeys, rq, rk, S);

    // 1024 rows, 8 waves/block -> 128 blocks
    topk_kernel<<<128, 256, 0, stream>>>(S, top_sim, idx);

    // 8192 prompt copies, one block each
    gather_kernel<<<BATCH * TOPK, 256, 0, stream>>>(prompts, idx, out);
}